// SlidingSelfAttention_19490561590104
// MI455X (gfx1250) — compile-verified
//
#include <hip/hip_runtime.h>

typedef float v2f __attribute__((ext_vector_type(2)));
typedef float v8f __attribute__((ext_vector_type(8)));

// D = A(16x4 f32) * B(4x16 f32) + C(16x16 f32), full f32 precision
#define WMMA_F32(a, b, c) \
  __builtin_amdgcn_wmma_f32_16x16x4_f32(false, (a), false, (b), (short)0, (c), false, false)

constexpr int Bb   = 2;
constexpr int Nn   = 2048;
constexpr int Ff   = 512;     // feature dim (GEMM K)
constexpr int Ww   = 128;     // window
constexpr int Dd   = 512;     // head dim
constexpr int E3   = 3 * Dd;  // 1536 (qkv row stride)
constexpr int ROWS = Bb * Nn; // 4096

// ---------------------------------------------------------------------------
// Kernel 1: qkv[m][e] = sum_k x[m][k] * W_lin[e][k]
// Both operands K-contiguous -> float2 loads match WMMA A/B lane layouts.
// Block: 256 threads = 8 waves in 4(M) x 2(N); wave does 32x64 = 2x4 tiles
// (2 A-loads + 4 B-loads feed 8 WMMAs per K-step -> load:WMMA = 0.75).
// Block tile: 128(M) x 128(N). Grid: (1536/128, 4096/128) = (12, 32).
// ---------------------------------------------------------------------------
__global__ __launch_bounds__(256) void qkv_gemm_kernel(
    const float* __restrict__ X, const float* __restrict__ Wl,
    float* __restrict__ QKV) {
  const int lane = threadIdx.x & 31;
  const int wave = threadIdx.x >> 5;
  const int c    = lane & 15;   // column within 16 (N for B/C, M for A)
  const int hi   = lane >> 4;   // half-wave select
  const int kb   = hi * 2;      // K sub-offset within the 4-wide step

  const int m0 = blockIdx.y * 128 + (wave >> 1) * 32;  // 4 waves in M
  const int n0 = blockIdx.x * 128 + (wave & 1) * 64;   // 2 waves in N

  const float* arow0 = X  + (size_t)(m0 + c)      * Ff + kb;
  const float* arow1 = X  + (size_t)(m0 + 16 + c) * Ff + kb;
  const float* brow0 = Wl + (size_t)(n0 + c)      * Ff + kb;
  const float* brow1 = Wl + (size_t)(n0 + 16 + c) * Ff + kb;
  const float* brow2 = Wl + (size_t)(n0 + 32 + c) * Ff + kb;
  const float* brow3 = Wl + (size_t)(n0 + 48 + c) * Ff + kb;

  v8f acc[2][4];
#pragma unroll
  for (int ti = 0; ti < 2; ++ti)
#pragma unroll
    for (int tj = 0; tj < 4; ++tj) acc[ti][tj] = v8f{};

#pragma unroll 4
  for (int k0 = 0; k0 < Ff; k0 += 4) {
    v2f a0 = *(const v2f*)(arow0 + k0);
    v2f a1 = *(const v2f*)(arow1 + k0);
    v2f b0 = *(const v2f*)(brow0 + k0);
    v2f b1 = *(const v2f*)(brow1 + k0);
    v2f b2 = *(const v2f*)(brow2 + k0);
    v2f b3 = *(const v2f*)(brow3 + k0);
    acc[0][0] = WMMA_F32(a0, b0, acc[0][0]);
    acc[0][1] = WMMA_F32(a0, b1, acc[0][1]);
    acc[0][2] = WMMA_F32(a0, b2, acc[0][2]);
    acc[0][3] = WMMA_F32(a0, b3, acc[0][3]);
    acc[1][0] = WMMA_F32(a1, b0, acc[1][0]);
    acc[1][1] = WMMA_F32(a1, b1, acc[1][1]);
    acc[1][2] = WMMA_F32(a1, b2, acc[1][2]);
    acc[1][3] = WMMA_F32(a1, b3, acc[1][3]);
  }

#pragma unroll
  for (int v = 0; v < 8; ++v) {
    const int mi = v + hi * 8;  // C/D layout: VGPR v holds M=v (lanes 0-15), M=v+8 (16-31)
#pragma unroll
    for (int ti = 0; ti < 2; ++ti) {
      float* orow = QKV + (size_t)(m0 + ti * 16 + mi) * E3 + n0 + c;
#pragma unroll
      for (int tj = 0; tj < 4; ++tj) orow[tj * 16] = acc[ti][tj][v];
    }
  }
}

// ---------------------------------------------------------------------------
// Kernel 2: sliding-window attention, one block per 16-query tile.
// Band of 9 key tiles. Phase1: S = Q.K^T (WMMA) + mask -> LDS.
// Phase2a: softmax with reference's zero-pad semantics.
// Phase2b: O = P.V (WMMA), each wave owns a 64-wide D slice.
// ---------------------------------------------------------------------------
__global__ __launch_bounds__(256) void sliding_attn_kernel(
    const float* __restrict__ QKV, float* __restrict__ Out) {
  __shared__ float S[16][148];  // 16 rows x 144 band cols (+pad)

  const int tid  = threadIdx.x;
  const int lane = tid & 31;
  const int wave = tid >> 5;
  const int c    = lane & 15;
  const int hi   = lane >> 4;
  const int kb   = hi * 2;

  const int tile  = blockIdx.x;        // 0..255
  const int r0    = tile * 16;         // global query row base (b*N + n0)
  const int bBase = (r0 / Nn) * Nn;    // batch row base
  const int n0    = r0 - bBase;        // query pos within batch
  const int tl    = n0 >> 4;           // local key-tile index of diagonal

  // ---- Phase 1: logits ----------------------------------------------------
  const float* qrow = QKV + (size_t)(r0 + c) * E3;  // q at cols [0,512)
  for (int i = wave; i < 9; i += 8) {
    const int kt = tl - 8 + i;
    if (kt >= 0) {
      const float* krow = QKV + (size_t)(bBase + kt * 16 + c) * E3 + Dd;
      v8f acc{};
#pragma unroll 4
      for (int k0 = 0; k0 < Dd; k0 += 4) {
        v2f a = *(const v2f*)(qrow + k0 + kb);
        v2f b = *(const v2f*)(krow + k0 + kb);
        acc = WMMA_F32(a, b, acc);
      }
#pragma unroll
      for (int v = 0; v < 8; ++v) {
        const int M = v + hi * 8;
        const int n = n0 + M;
        const int m = kt * 16 + c;
        const bool valid = (m <= n) && (m >= n - (Ww - 1));
        S[M][i * 16 + c] = valid ? acc[v] : -__builtin_inff();
      }
    } else {
#pragma unroll
      for (int v = 0; v < 8; ++v) S[v + hi * 8][i * 16 + c] = -__builtin_inff();
    }
  }
  __syncthreads();

  // ---- Phase 2a: softmax (ref pads k,v with zeros: pad logits are 0) ------
  if (tid < 16) {
    const int n = n0 + tid;
    int pads = (Ww - 1) - n;
    if (pads < 0) pads = 0;
    float mx = (pads > 0) ? 0.f : -__builtin_inff();
    for (int j = 0; j < 144; ++j) mx = fmaxf(mx, S[tid][j]);
    float denom = (float)pads * __expf(0.f - mx);
    for (int j = 0; j < 144; ++j) {
      const float e = __expf(S[tid][j] - mx);  // exp(-inf)=0 kills masked slots
      S[tid][j] = e;
      denom += e;
    }
    const float inv = 1.f / denom;
    for (int j = 0; j < 144; ++j) S[tid][j] *= inv;
  }
  __syncthreads();

  // ---- Phase 2b: out = P @ V ---------------------------------------------
  const int d0 = wave * 64;  // 8 waves x 64 = 512
  v8f o[4] = {v8f{}, v8f{}, v8f{}, v8f{}};

  for (int i = 0; i < 9; ++i) {
    const int kt = tl - 8 + i;
    if (kt < 0) continue;  // pad region: v == 0, contributes nothing
#pragma unroll
    for (int ks = 0; ks < 16; ks += 4) {
      // A = P[M = lane&15][k = i*16+ks+kb .. +1], even col -> 8B aligned
      v2f a = *(const v2f*)(&S[c][i * 16 + ks + kb]);
      const float* vr0 = QKV + (size_t)(bBase + kt * 16 + ks + kb) * E3 + 2 * Dd + d0;
      const float* vr1 = vr0 + E3;
#pragma unroll
      for (int g = 0; g < 4; ++g) {
        v2f b;
        b.x = vr0[g * 16 + c];  // B[k  ][d] = V[k  ][d0+g*16+c]
        b.y = vr1[g * 16 + c];  // B[k+1][d]
        o[g] = WMMA_F32(a, b, o[g]);
      }
    }
  }

#pragma unroll
  for (int v = 0; v < 8; ++v) {
    const int M = v + hi * 8;
    float* orow = Out + (size_t)(r0 + M) * Dd + d0;
#pragma unroll
    for (int g = 0; g < 4; ++g) orow[g * 16 + c] = o[g][v];
  }
}

// ---------------------------------------------------------------------------
extern "C" void kernel_launch(void* const* d_in, const int* in_sizes, int n_in,
                              void* d_out, int out_size, void* d_ws, size_t ws_size,
                              hipStream_t stream) {
  (void)in_sizes; (void)n_in; (void)out_size; (void)ws_size;
  const float* x   = (const float*)d_in[0];  // (2,2048,512) f32
  const float* Wl  = (const float*)d_in[1];  // (1536,512)   f32
  float*       out = (float*)d_out;          // (2,2048,512) f32
  float*       qkv = (float*)d_ws;           // 4096 x 1536 f32 scratch (25.2 MB)

  dim3 gA(E3 / 128, ROWS / 128);  // (12, 32)
  qkv_gemm_kernel<<<gA, 256, 0, stream>>>(x, Wl, qkv);
  sliding_attn_kernel<<<ROWS / 16, 256, 0, stream>>>(qkv, out);
}